// FarthestPointSampler_36283883717166
// MI455X (gfx1250) — compile-verified
//
#include <hip/hip_runtime.h>

// Farthest Point Sampling for MI455X (gfx1250, wave32).
// B=64 clouds, N=65536 pts x 3 f32, 1024 sequential selections per cloud.
// One workgroup (1024 threads = 32 waves) per cloud; mindist lives in LDS
// (256KB of the 320KB/WGP CDNA5 LDS); coords stay resident in the 192MB L2
// (48MB total working set, warmed with global_prefetch_b8). The per-iteration
// query-point broadcast uses GLOBAL_LOAD_ASYNC_TO_LDS_B32 (ASYNCcnt) to skip
// the VGPR round-trip on the serial critical path.

#define NPTS      65536
#define NSAMP     1024
#define NTHREADS  1024
#define PER_THREAD (NPTS / NTHREADS)   // 64 points per thread
#define NWAVES    (NTHREADS / 32)      // 32 waves (wave32)

struct __align__(4) F3 { float x, y, z; };

// dynamic LDS layout: [ mindist: NPTS floats | wdist: 32 f | widx: 32 i | q: 3 f | pad ]
#define SMEM_BYTES (NPTS * 4 + NWAVES * 4 + NWAVES * 4 + 4 * 4)

// Async DMA of one 12-byte point (4B-aligned -> three B32 transfers) from
// global memory directly into LDS, bypassing VGPRs. Caller must be inside a
// single-lane-active region (we call it under `if (t == 0)`), and the
// following s_wait_asynccnt 0 guarantees LDS contents before the barrier.
__device__ __forceinline__ void async_point_to_lds(const float* __restrict__ gsrc,
                                                   float* lds_dst) {
    // Low 32 bits of a generic pointer into LDS == wave-relative LDS byte addr.
    unsigned ldsa = (unsigned)(uintptr_t)lds_dst;
    asm volatile("global_load_async_to_lds_b32 %0, %1, off"
                 :: "v"(ldsa), "v"(gsrc) : "memory");
    asm volatile("global_load_async_to_lds_b32 %0, %1, off offset:4"
                 :: "v"(ldsa), "v"(gsrc) : "memory");
    asm volatile("global_load_async_to_lds_b32 %0, %1, off offset:8"
                 :: "v"(ldsa), "v"(gsrc) : "memory");
    asm volatile("s_wait_asynccnt 0x0" ::: "memory");
}

__global__ __launch_bounds__(NTHREADS)
void fps_kernel(const float* __restrict__ pos,
                const int*   __restrict__ start_idx,
                int*         __restrict__ out) {
    extern __shared__ char smem_raw[];
    float* mindist = reinterpret_cast<float*>(smem_raw);
    float* wdist   = reinterpret_cast<float*>(smem_raw + NPTS * 4);
    int*   widx    = reinterpret_cast<int*>(wdist + NWAVES);
    float* qs      = reinterpret_cast<float*>(widx + NWAVES);

    const int b = blockIdx.x;
    const int t = threadIdx.x;
    const float* __restrict__ posb = pos + (size_t)b * NPTS * 3;
    const F3* __restrict__ p = reinterpret_cast<const F3*>(posb);
    int* __restrict__ outb = out + (size_t)b * NSAMP;

    int cur = start_idx[b];   // only meaningful on thread 0 after the loop starts

    // Init running min-distance to +inf and warm L2 with this cloud's coords.
    for (int k = 0; k < PER_THREAD; ++k) {
        const int idx = t + (k << 10);
        mindist[idx] = __builtin_inff();
        __builtin_prefetch(&p[idx], 0, 1);   // -> global_prefetch_b8 scope:DEV
    }
    if (t == 0) {
        async_point_to_lds(posb + (size_t)cur * 3, qs);  // publish q(cur) via DMA
    }
    __syncthreads();

    for (int i = 0; i < NSAMP; ++i) {
        if (t == 0) outb[i] = cur;

        const float qx = qs[0], qy = qs[1], qz = qs[2];
        float best    = -1.0f;   // mindist >= 0 always
        int   bestIdx = t;

        // Distance pass: strided mapping keeps per-instruction lane addresses
        // contiguous (coalesced 12B/lane b96 loads, all L2 hits after warmup).
        #pragma unroll 16
        for (int k = 0; k < PER_THREAD; ++k) {
            const int idx = t + (k << 10);
            F3 v = p[idx];
            const float dx = v.x - qx;
            const float dy = v.y - qy;
            const float dz = v.z - qz;
            const float d  = dx * dx + dy * dy + dz * dz;
            float m = mindist[idx];
            m = fminf(m, d);
            mindist[idx] = m;
            // strict '>' keeps the smallest idx within this thread (ascending idx)
            if (m > best) { best = m; bestIdx = idx; }
        }

        // wave32 butterfly argmax (dist desc, idx asc on ties) -> ds permutes
        #pragma unroll
        for (int off = 16; off > 0; off >>= 1) {
            const float ob = __shfl_xor(best, off, 32);
            const int   oi = __shfl_xor(bestIdx, off, 32);
            if (ob > best || (ob == best && oi < bestIdx)) { best = ob; bestIdx = oi; }
        }
        const int wid = t >> 5;
        if ((t & 31) == 0) { wdist[wid] = best; widx[wid] = bestIdx; }
        __syncthreads();

        // wave 0 reduces the 32 per-wave partials (one lane each)
        if (t < 32) {
            best = wdist[t]; bestIdx = widx[t];
            #pragma unroll
            for (int off = 16; off > 0; off >>= 1) {
                const float ob = __shfl_xor(best, off, 32);
                const int   oi = __shfl_xor(bestIdx, off, 32);
                if (ob > best || (ob == best && oi < bestIdx)) { best = ob; bestIdx = oi; }
            }
            if (t == 0) {
                cur = bestIdx;                                   // next selection
                async_point_to_lds(posb + (size_t)cur * 3, qs);  // DMA next q into LDS
            }
        }
        __syncthreads();   // q ready; also protects wdist/widx WAR for next iter
    }
}

extern "C" void kernel_launch(void* const* d_in, const int* in_sizes, int n_in,
                              void* d_out, int out_size, void* d_ws, size_t ws_size,
                              hipStream_t stream) {
    const float* pos       = reinterpret_cast<const float*>(d_in[0]);
    const int*   start_idx = reinterpret_cast<const int*>(d_in[1]);
    int*         out       = reinterpret_cast<int*>(d_out);

    const int B = in_sizes[1];   // 64 clouds

    // 256KB+ dynamic LDS: raise the per-kernel cap (deterministic, capture-safe).
    (void)hipFuncSetAttribute(reinterpret_cast<const void*>(fps_kernel),
                              hipFuncAttributeMaxDynamicSharedMemorySize, SMEM_BYTES);

    fps_kernel<<<B, NTHREADS, SMEM_BYTES, stream>>>(pos, start_idx, out);

    (void)n_in; (void)out_size; (void)d_ws; (void)ws_size;
}